// GateGRU_74148315398656
// MI455X (gfx1250) — compile-verified
//
#include <hip/hip_runtime.h>

typedef __attribute__((ext_vector_type(2))) float v2f;
typedef __attribute__((ext_vector_type(8))) float v8f;

#define N_BATCH 128
#define T_STEPS 512
#define H_DIM   5
#define KPAD    8
#define NEURONS 2000
#define M_ROWS  (N_BATCH * T_STEPS)   // 65536
#define O_TILES (NEURONS / 16)        // 125
#define M_TILES (M_ROWS / 16)         // 4096
#define TILES_PER_WAVE 16
#define M_STRIPS (M_TILES / TILES_PER_WAVE)  // 256

// ---------------------------------------------------------------------------
// Phase 1: GRU scan. One wave (32 lanes) per batch element. Lanes 0..14 each
// own one of the 3H=15 gate rows; h is kept distributed (lane j<5 owns h_j)
// and broadcast each step via wave shuffles. States are written to d_ws as a
// [M_ROWS, 8] row-major fp32 matrix, zero-padded in cols 5..7 so phase 2 can
// feed V_WMMA_F32_16X16X4_F32 with K padded to 8.
// ---------------------------------------------------------------------------
__global__ __launch_bounds__(32) void gru_scan_kernel(
    const float* __restrict__ x,      // [N, T, 3]
    const float* __restrict__ w_ih,   // [15, 3]
    const float* __restrict__ w_hh,   // [15, 5]
    const float* __restrict__ b_ih,   // [15]
    const float* __restrict__ b_hh,   // [15]
    float* __restrict__ states)       // [N*T, 8]
{
    const int n    = blockIdx.x;
    const int lane = threadIdx.x;     // 0..31, wave32

    // Per-lane gate weights (lanes 0..14 meaningful).
    float wi0 = 0.f, wi1 = 0.f, wi2 = 0.f;
    float wh0 = 0.f, wh1 = 0.f, wh2 = 0.f, wh3 = 0.f, wh4 = 0.f;
    float bi = 0.f, bh = 0.f;
    if (lane < 15) {
        wi0 = w_ih[lane * 3 + 0];
        wi1 = w_ih[lane * 3 + 1];
        wi2 = w_ih[lane * 3 + 2];
        wh0 = w_hh[lane * 5 + 0];
        wh1 = w_hh[lane * 5 + 1];
        wh2 = w_hh[lane * 5 + 2];
        wh3 = w_hh[lane * 5 + 3];
        wh4 = w_hh[lane * 5 + 4];
        bi  = b_ih[lane];
        bh  = b_hh[lane];
    }

    float hd = 0.f;                   // lane j<5 holds h_j
    const float* xp = x + (size_t)n * T_STEPS * 3;
    float* sp = states + (size_t)n * T_STEPS * KPAD;

    for (int t = 0; t < T_STEPS; ++t) {
        // Broadcast hidden state to all lanes.
        const float h0 = __shfl(hd, 0, 32);
        const float h1 = __shfl(hd, 1, 32);
        const float h2 = __shfl(hd, 2, 32);
        const float h3 = __shfl(hd, 3, 32);
        const float h4 = __shfl(hd, 4, 32);

        const float x0 = xp[0], x1 = xp[1], x2 = xp[2];
        xp += 3;

        // gi = b_ih + x . w_ih[row]  ;  gh = b_hh + h . w_hh[row]
        const float gi = fmaf(x2, wi2, fmaf(x1, wi1, fmaf(x0, wi0, bi)));
        const float gh = fmaf(h4, wh4, fmaf(h3, wh3,
                         fmaf(h2, wh2, fmaf(h1, wh1, fmaf(h0, wh0, bh)))));

        // r/z lanes: sigmoid(gi+gh); computed in every lane (harmless).
        const float s  = 1.0f / (1.0f + __expf(-(gi + gh)));
        // n lanes (10..14): need r from lanes 0..4 -> lane%5 maps 10..14 -> 0..4.
        const float rr = __shfl(s, lane % 5, 32);
        const float nn = tanhf(fmaf(rr, gh, gi));

        // Update lanes 0..4: h_new = (1-z)*n + z*h
        const float zj = __shfl(s,  lane + 5,  32);   // z_j from lanes 5..9
        const float nj = __shfl(nn, lane + 10, 32);   // n_j from lanes 10..14
        const float hnew = fmaf(zj, hd, (1.0f - zj) * nj);
        if (lane < 5) hd = hnew;

        // Store row m = n*T + t with K padded to 8 (cols 5..7 zero).
        if (lane < KPAD) {
            sp[lane] = (lane < H_DIM) ? __shfl(hnew, lane, 32) : 0.0f;
        }
        sp += KPAD;
    }
}

// ---------------------------------------------------------------------------
// Phase 2: out[m, o] = exp(states[m, 0:5] . lw[o, 0:5] + lb[o])
// One wave per 16-wide O tile; wave loads B (lw^T tile, K padded to 8) and
// the bias once, then sweeps 16 M tiles, 2 chained v_wmma_f32_16x16x4_f32
// per tile. Store-bandwidth bound (524 MB out, never re-read) -> use
// non-temporal stores so the 2 MB states matrix + lw stay hot in L2 across
// the 125 reuse passes.
// ---------------------------------------------------------------------------
__global__ __launch_bounds__(256) void proj_exp_kernel(
    const float* __restrict__ A,   // [M_ROWS, 8] padded states
    const float* __restrict__ lw,  // [NEURONS, 5]
    const float* __restrict__ lb,  // [NEURONS]
    float* __restrict__ out)       // [M_ROWS, NEURONS]
{
    const int waveInBlock = threadIdx.x >> 5;
    const int lane        = threadIdx.x & 31;
    const int waveId      = blockIdx.x * 8 + waveInBlock;

    const int oTile  = waveId % O_TILES;            // 0..124
    const int mStrip = waveId / O_TILES;            // 0..255

    const int col  = (lane & 15);                   // N index within tile
    const int oo   = oTile * 16 + col;
    const bool lo  = (lane < 16);

    // B operand: 4x16 tile of lw^T. VGPR0 holds K=(lo?0:2), VGPR1 K=(lo?1:3).
    const float* lwp = lw + oo * H_DIM;
    v2f b0, b1;
    b0.x = lo ? lwp[0] : lwp[2];
    b0.y = lo ? lwp[1] : lwp[3];
    b1.x = lo ? lwp[4] : 0.0f;      // K=4 real; K=5,6,7 padded zero
    b1.y = 0.0f;

    const float bias = lb[oo];

    const int aKoff = lo ? 0 : 2;   // A: VGPR0 K=(lo?0:2), VGPR1 K=(lo?1:3)

    for (int mi = 0; mi < TILES_PER_WAVE; ++mi) {
        const int mBase = (mStrip * TILES_PER_WAVE + mi) * 16;
        const float* ap = A + (size_t)(mBase + col) * KPAD;

        const v2f a0 = *(const v2f*)(ap + aKoff);       // K 0..3
        const v2f a1 = *(const v2f*)(ap + 4 + aKoff);   // K 4..7 (zero padded)

        v8f c = {};
        c = __builtin_amdgcn_wmma_f32_16x16x4_f32(
                false, a0, false, b0, (short)0, c, false, false);
        c = __builtin_amdgcn_wmma_f32_16x16x4_f32(
                false, a1, false, b1, (short)0, c, false, false);

        // D layout: VGPR v -> row M = v + (lane>=16 ? 8 : 0), col = lane&15.
        const int rowAdd = lo ? 0 : 8;
        float* op = out + (size_t)(mBase + rowAdd) * NEURONS + oo;
        #pragma unroll
        for (int v = 0; v < 8; ++v) {
            __builtin_nontemporal_store(__expf(c[v] + bias), op + (size_t)v * NEURONS);
        }
    }
}

extern "C" void kernel_launch(void* const* d_in, const int* in_sizes, int n_in,
                              void* d_out, int out_size, void* d_ws, size_t ws_size,
                              hipStream_t stream) {
    const float* x    = (const float*)d_in[0];
    const float* w_ih = (const float*)d_in[1];
    const float* w_hh = (const float*)d_in[2];
    const float* b_ih = (const float*)d_in[3];
    const float* b_hh = (const float*)d_in[4];
    const float* lw   = (const float*)d_in[5];
    const float* lb   = (const float*)d_in[6];
    float* out    = (float*)d_out;
    float* states = (float*)d_ws;   // needs M_ROWS * 8 * 4 = 2 MB

    gru_scan_kernel<<<N_BATCH, 32, 0, stream>>>(x, w_ih, w_hh, b_ih, b_hh, states);

    const int totalWaves = M_STRIPS * O_TILES;      // 256 * 125 = 32000
    proj_exp_kernel<<<totalWaves / 8, 256, 0, stream>>>(states, lw, lb, out);
}